// SmoothGAT_19155554140400
// MI455X (gfx1250) — compile-verified
//
#include <hip/hip_runtime.h>
#include <math.h>

typedef __attribute__((ext_vector_type(2))) float v2f;
typedef __attribute__((ext_vector_type(8))) float v8f;

#define NEG_SLOPE 0.2f

// ---------------------------------------------------------------------------
// float atomic max via int/uint ordering trick (works for all finite + -inf)
// ---------------------------------------------------------------------------
__device__ __forceinline__ void atomicMaxF32(float* addr, float v) {
    if (v >= 0.0f) {
        atomicMax((int*)addr, __float_as_int(v));
    } else {
        atomicMin((unsigned int*)addr, __float_as_uint(v));
    }
}

// ---------------------------------------------------------------------------
// GEMM1: H[N x 64] = X[N x 128] @ W[128 x 64], fp32 WMMA 16x16x4
// block = 128 threads (4 waves); wave w computes cols [16w, 16w+16) of a
// 16-row tile; grid.x = ceil(N/16)
// ---------------------------------------------------------------------------
__global__ void gat_gemm1_wmma(const float* __restrict__ X,
                               const float* __restrict__ W,
                               float* __restrict__ H, int N) {
    const int wave = threadIdx.x >> 5;
    const int lane = threadIdx.x & 31;
    const int half = lane >> 4;      // 0: lanes 0-15, 1: lanes 16-31
    const int l16  = lane & 15;
    const int row0 = blockIdx.x * 16;
    const int col  = wave * 16 + l16;
    const int arow = min(row0 + l16, N - 1);
    const float* __restrict__ xrow = X + (size_t)arow * 128;

    v8f acc = {};
    #pragma unroll 8
    for (int k = 0; k < 128; k += 4) {
        const int ka = k + half * 2;
        v2f a, b;
        a.x = xrow[ka];
        a.y = xrow[ka + 1];
        b.x = W[ka * 64 + col];
        b.y = W[(ka + 1) * 64 + col];
        acc = __builtin_amdgcn_wmma_f32_16x16x4_f32(
            false, a, false, b, (short)0, acc, false, false);
    }
    const int mbase = row0 + half * 8;
    if (row0 + 16 <= N) {
        // fast path: full tile, no per-row exec masking
        #pragma unroll
        for (int r = 0; r < 8; ++r)
            H[(size_t)(mbase + r) * 64 + col] = acc[r];
    } else {
        #pragma unroll
        for (int r = 0; r < 8; ++r) {
            const int m = mbase + r;
            if (m < N) H[(size_t)m * 64 + col] = acc[r];
        }
    }
}

// ---------------------------------------------------------------------------
// GEMM2: H[N x 40] = X[N x 64] @ W[64 x 40], fp32 WMMA, cols padded to 48
// block = 96 threads (3 waves)
// ---------------------------------------------------------------------------
__global__ void gat_gemm2_wmma(const float* __restrict__ X,
                               const float* __restrict__ W,
                               float* __restrict__ H, int N) {
    const int wave = threadIdx.x >> 5;
    const int lane = threadIdx.x & 31;
    const int half = lane >> 4;
    const int l16  = lane & 15;
    const int row0 = blockIdx.x * 16;
    const int col  = wave * 16 + l16;
    const int arow = min(row0 + l16, N - 1);
    const float* __restrict__ xrow = X + (size_t)arow * 64;
    const bool colok = (col < 40);

    v8f acc = {};
    #pragma unroll 4
    for (int k = 0; k < 64; k += 4) {
        const int ka = k + half * 2;
        v2f a, b;
        a.x = xrow[ka];
        a.y = xrow[ka + 1];
        if (colok) {
            b.x = W[ka * 40 + col];
            b.y = W[(ka + 1) * 40 + col];
        } else {
            b.x = 0.0f;
            b.y = 0.0f;
        }
        acc = __builtin_amdgcn_wmma_f32_16x16x4_f32(
            false, a, false, b, (short)0, acc, false, false);
    }
    if (colok) {
        const int mbase = row0 + half * 8;
        if (row0 + 16 <= N) {
            #pragma unroll
            for (int r = 0; r < 8; ++r)
                H[(size_t)(mbase + r) * 40 + col] = acc[r];
        } else {
            #pragma unroll
            for (int r = 0; r < 8; ++r) {
                const int m = mbase + r;
                if (m < N) H[(size_t)m * 40 + col] = acc[r];
            }
        }
    }
}

// ---------------------------------------------------------------------------
// per-node attention logits: als[i] = <h[i], a_src>, ald[i] = <h[i], a_dst>
// F must be a multiple of 4 (64 and 40 both are)
// ---------------------------------------------------------------------------
__global__ void attn_logits_k(const float* __restrict__ h,
                              const float* __restrict__ asrc,
                              const float* __restrict__ adst,
                              float* __restrict__ als,
                              float* __restrict__ ald, int N, int F) {
    const int i = blockIdx.x * blockDim.x + threadIdx.x;
    if (i >= N) return;
    const float4* __restrict__ row = (const float4*)(h + (size_t)i * F);
    const float4* __restrict__ a4  = (const float4*)asrc;
    const float4* __restrict__ d4  = (const float4*)adst;
    float s = 0.0f, d = 0.0f;
    const int F4 = F >> 2;
    for (int c = 0; c < F4; ++c) {
        const float4 v  = row[c];
        const float4 va = a4[c];
        const float4 vd = d4[c];
        s += v.x * va.x + v.y * va.y + v.z * va.z + v.w * va.w;
        d += v.x * vd.x + v.y * vd.y + v.z * vd.z + v.w * vd.w;
    }
    als[i] = s;
    ald[i] = d;
}

__global__ void init_mz_k(float* __restrict__ m, float* __restrict__ z, int N) {
    const int i = blockIdx.x * blockDim.x + threadIdx.x;
    if (i >= N) return;
    m[i] = -INFINITY;
    z[i] = 0.0f;
}

__global__ void zero_f32_k(float* __restrict__ p, long long n) {
    const long long i = (long long)blockIdx.x * blockDim.x + threadIdx.x;
    if (i < n) p[i] = 0.0f;
}

// ---------------------------------------------------------------------------
// edge pass 1: segment max of leaky-relu'd logits over dst
// edges [0,E): from edge_index; edges [E,E+N): self loops
// ---------------------------------------------------------------------------
__global__ void edge_max_k(const int* __restrict__ ei, int E, int N,
                           const float* __restrict__ als,
                           const float* __restrict__ ald,
                           float* __restrict__ m) {
    const int e = blockIdx.x * blockDim.x + threadIdx.x;
    if (e >= E + N) return;
    int s, d;
    if (e < E) { s = ei[e]; d = ei[E + e]; }
    else       { s = d = e - E; }
    float v = als[s] + ald[d];
    v = (v > 0.0f) ? v : NEG_SLOPE * v;
    atomicMaxF32(&m[d], v);
}

// ---------------------------------------------------------------------------
// edge pass 2: w = exp(e - m[dst]); cache per-edge w; z[dst] += w
// ---------------------------------------------------------------------------
__global__ void edge_sum_k(const int* __restrict__ ei, int E, int N,
                           const float* __restrict__ als,
                           const float* __restrict__ ald,
                           const float* __restrict__ m,
                           float* __restrict__ z,
                           float* __restrict__ wbuf) {
    const int e = blockIdx.x * blockDim.x + threadIdx.x;
    if (e >= E + N) return;
    int s, d;
    if (e < E) { s = ei[e]; d = ei[E + e]; }
    else       { s = d = e - E; }
    float v = als[s] + ald[d];
    v = (v > 0.0f) ? v : NEG_SLOPE * v;
    const float w = __expf(v - m[d]);
    wbuf[e] = w;
    atomicAdd(&z[d], w);
}

// ---------------------------------------------------------------------------
// edge pass 3: out[dst] += h[src] * (w / (z[dst]+1e-16)); 4 channels/thread.
// CH (chunks per edge) is a compile-time constant so the e = t/CH, c = t%CH
// decomposition strength-reduces (no runtime integer division).
// ---------------------------------------------------------------------------
template <int CH>
__global__ void edge_aggr_k(const int* __restrict__ ei, int E, int N,
                            const float* __restrict__ wbuf,
                            const float* __restrict__ z,
                            const float* __restrict__ h,
                            float* __restrict__ out) {
    constexpr int F = CH * 4;
    const long long t = (long long)blockIdx.x * blockDim.x + threadIdx.x;
    const long long tot = (long long)(E + N) * CH;
    if (t >= tot) return;
    const int e  = (int)(t / CH);
    const int c4 = (int)(t % CH) * 4;
    int s, d;
    if (e < E) { s = ei[e]; d = ei[E + e]; }
    else       { s = d = e - E; }
    const float alpha = wbuf[e] / (z[d] + 1e-16f);
    const float4 hv = *(const float4*)(h + (size_t)s * F + c4);
    float* o = out + (size_t)d * F + c4;
    atomicAdd(o + 0, hv.x * alpha);
    atomicAdd(o + 1, hv.y * alpha);
    atomicAdd(o + 2, hv.z * alpha);
    atomicAdd(o + 3, hv.w * alpha);
}

// ---------------------------------------------------------------------------
// h1r = relu(agg1 + b1)
// ---------------------------------------------------------------------------
__global__ void bias_relu_k(const float* __restrict__ agg,
                            const float* __restrict__ b,
                            float* __restrict__ out, long long n, int F) {
    const long long i = (long long)blockIdx.x * blockDim.x + threadIdx.x;
    if (i >= n) return;
    const float v = agg[i] + b[(int)(i % F)];
    out[i] = (v > 0.0f) ? v : 0.0f;
}

// ---------------------------------------------------------------------------
// log_softmax over 40 channels per node: out = v - max - log(sum exp(v-max))
// ---------------------------------------------------------------------------
__global__ void log_softmax_k(const float* __restrict__ agg,
                              const float* __restrict__ b,
                              float* __restrict__ out, int N) {
    const int i = blockIdx.x * blockDim.x + threadIdx.x;
    if (i >= N) return;
    float v[40];
    float mx = -INFINITY;
    #pragma unroll
    for (int c = 0; c < 40; ++c) {
        v[c] = agg[(size_t)i * 40 + c] + b[c];
        mx = fmaxf(mx, v[c]);
    }
    float s = 0.0f;
    #pragma unroll
    for (int c = 0; c < 40; ++c) s += __expf(v[c] - mx);
    const float l = __logf(s);
    #pragma unroll
    for (int c = 0; c < 40; ++c) out[(size_t)i * 40 + c] = v[c] - mx - l;
}

// ---------------------------------------------------------------------------
extern "C" void kernel_launch(void* const* d_in, const int* in_sizes, int n_in,
                              void* d_out, int out_size, void* d_ws, size_t ws_size,
                              hipStream_t stream) {
    const float* x      = (const float*)d_in[0];
    const int*   ei     = (const int*)  d_in[1];   // [2, E] int32: row0=src, row1=dst
    const float* W1     = (const float*)d_in[2];
    const float* a_src1 = (const float*)d_in[3];
    const float* a_dst1 = (const float*)d_in[4];
    const float* b1     = (const float*)d_in[5];
    const float* W2     = (const float*)d_in[6];
    const float* a_src2 = (const float*)d_in[7];
    const float* a_dst2 = (const float*)d_in[8];
    const float* b2     = (const float*)d_in[9];
    float* out = (float*)d_out;

    const int N  = in_sizes[0] / 128;   // 100000
    const int E  = in_sizes[1] / 2;     // 1600000
    const int ET = E + N;               // + self loops

    // workspace layout (floats)
    float* h1   = (float*)d_ws;               // N*64  (later overwritten by relu'd h1)
    float* agg1 = h1   + (size_t)N * 64;      // N*64
    float* h2   = agg1 + (size_t)N * 64;      // N*40
    float* agg2 = h2   + (size_t)N * 40;      // N*40
    float* als  = agg2 + (size_t)N * 40;      // N
    float* ald  = als  + N;                   // N
    float* mbuf = ald  + N;                   // N
    float* zbuf = mbuf + N;                   // N
    float* wbuf = zbuf + N;                   // ET

    const int TB = 256;
    const int nb_nodes = (N + TB - 1) / TB;
    const int nb_edges = (ET + TB - 1) / TB;

    // ---------------- Layer 1 ----------------
    gat_gemm1_wmma<<<(N + 15) / 16, 128, 0, stream>>>(x, W1, h1, N);
    attn_logits_k<<<nb_nodes, TB, 0, stream>>>(h1, a_src1, a_dst1, als, ald, N, 64);
    init_mz_k<<<nb_nodes, TB, 0, stream>>>(mbuf, zbuf, N);
    {
        const long long n = (long long)N * 64;
        zero_f32_k<<<(unsigned)((n + TB - 1) / TB), TB, 0, stream>>>(agg1, n);
    }
    edge_max_k<<<nb_edges, TB, 0, stream>>>(ei, E, N, als, ald, mbuf);
    edge_sum_k<<<nb_edges, TB, 0, stream>>>(ei, E, N, als, ald, mbuf, zbuf, wbuf);
    {
        const long long tot = (long long)ET * 16;   // 64 ch / 4 per thread
        edge_aggr_k<16><<<(unsigned)((tot + TB - 1) / TB), TB, 0, stream>>>(
            ei, E, N, wbuf, zbuf, h1, agg1);
    }
    {
        const long long n = (long long)N * 64;
        bias_relu_k<<<(unsigned)((n + TB - 1) / TB), TB, 0, stream>>>(agg1, b1, h1, n, 64);
    }

    // ---------------- Layer 2 ----------------
    gat_gemm2_wmma<<<(N + 15) / 16, 96, 0, stream>>>(h1, W2, h2, N);
    attn_logits_k<<<nb_nodes, TB, 0, stream>>>(h2, a_src2, a_dst2, als, ald, N, 40);
    init_mz_k<<<nb_nodes, TB, 0, stream>>>(mbuf, zbuf, N);
    {
        const long long n = (long long)N * 40;
        zero_f32_k<<<(unsigned)((n + TB - 1) / TB), TB, 0, stream>>>(agg2, n);
    }
    edge_max_k<<<nb_edges, TB, 0, stream>>>(ei, E, N, als, ald, mbuf);
    edge_sum_k<<<nb_edges, TB, 0, stream>>>(ei, E, N, als, ald, mbuf, zbuf, wbuf);
    {
        const long long tot = (long long)ET * 10;   // 40 ch / 4 per thread
        edge_aggr_k<10><<<(unsigned)((tot + TB - 1) / TB), TB, 0, stream>>>(
            ei, E, N, wbuf, zbuf, h2, agg2);
    }
    log_softmax_k<<<nb_nodes, TB, 0, stream>>>(agg2, b2, out, N);
}